// RKDLoss_43215960932890
// MI455X (gfx1250) — compile-verified
//
#include <hip/hip_runtime.h>
#include <hip/hip_bf16.h>

#define N 384
#define M 3
#define DD 256
#define NNEG 16
#define BIGV 1.0e9f

typedef __attribute__((ext_vector_type(2))) float v2f;
typedef __attribute__((ext_vector_type(8))) float v8f;

__device__ __forceinline__ float huber_f(float x, float y) {
    float d = fabsf(x - y);
    return (d <= 1.0f) ? 0.5f * d * d : d - 0.5f;
}
__device__ __forceinline__ float safe_div_f(float a, float b) {
    return (b == 0.0f) ? 0.0f : a / b;
}
__device__ __forceinline__ float wred(float v) {
#pragma unroll
    for (int off = 16; off > 0; off >>= 1) v += __shfl_xor(v, off, 32);
    return v;
}

// --- 1. se = mean(src over m); squared norms for emb, se, src0..2 ------------
__global__ void prep_kernel(const float* __restrict__ emb, const float* __restrict__ src,
                            float* __restrict__ se, float* __restrict__ sq) {
    int i = blockIdx.x * blockDim.x + threadIdx.x;
    if (i >= N) return;
    float qe = 0.f, qs = 0.f, q0 = 0.f, q1 = 0.f, q2 = 0.f;
    for (int j = 0; j < DD; ++j) {
        float e = emb[i * DD + j];
        qe += e * e;
        float a = src[(i * M + 0) * DD + j];
        float b = src[(i * M + 1) * DD + j];
        float c = src[(i * M + 2) * DD + j];
        q0 += a * a; q1 += b * b; q2 += c * c;
        float mv = (a + b + c) * (1.0f / (float)M);
        se[i * DD + j] = mv;
        qs += mv * mv;
    }
    sq[0 * N + i] = qe;
    sq[1 * N + i] = qs;
    sq[2 * N + i] = q0;
    sq[3 * N + i] = q1;
    sq[4 * N + i] = q2;
}

// --- 2. Gram via V_WMMA_F32_16X16X4_F32 -> euclidean distance matrices ------
// blockIdx.y: 0 = emb->D_emb, 1 = se->D_se, 2 = avg over 3 src sets -> savg
__global__ void gram_dist_kernel(const float* __restrict__ emb, const float* __restrict__ src,
                                 const float* __restrict__ se, const float* __restrict__ sq,
                                 float* __restrict__ Demb, float* __restrict__ Dse,
                                 float* __restrict__ savg) {
    const int lane = threadIdx.x & 31;
    const int wave = threadIdx.x >> 5;
    const int TB = N / 16;                       // 24 tiles per dim
    const int tile = blockIdx.x * 4 + wave;      // 144*4 == 576 == TB*TB exactly
    if (tile >= TB * TB) return;
    const int bi = tile / TB, bj = tile % TB;
    const int half = lane >> 4, mrow = lane & 15;
    const int rowA = bi * 16 + mrow;
    const int rowB = bj * 16 + mrow;
    const int matId = blockIdx.y;

    float dsum[8];
#pragma unroll
    for (int r = 0; r < 8; ++r) dsum[r] = 0.0f;

    const int nsrc = (matId == 2) ? M : 1;
    for (int s = 0; s < nsrc; ++s) {
        const float* X;
        const float* sqv;
        int stride;
        if (matId == 0)      { X = emb;         sqv = sq + 0 * N;       stride = DD; }
        else if (matId == 1) { X = se;          sqv = sq + 1 * N;       stride = DD; }
        else                 { X = src + s*DD;  sqv = sq + (2 + s) * N; stride = M * DD; }
        const float* pa = X + (size_t)rowA * stride + 2 * half;
        const float* pb = X + (size_t)rowB * stride + 2 * half;
        v8f c = {};
        for (int k = 0; k < DD; k += 4) {
            v2f a, b;
            a.x = pa[k]; a.y = pa[k + 1];
            b.x = pb[k]; b.y = pb[k + 1];
            // D = A(16x4,f32) * B(4x16,f32) + C  -> v_wmma_f32_16x16x4_f32
            c = __builtin_amdgcn_wmma_f32_16x16x4_f32(false, a, false, b,
                                                      (short)0, c, false, false);
        }
#pragma unroll
        for (int r = 0; r < 8; ++r) {
            int row = bi * 16 + r + 8 * half;
            float d2 = sqv[row] + sqv[rowB] - 2.0f * c[r];
            float dd = sqrtf(fmaxf(d2, 0.0f));
            if (row == rowB) dd = 0.0f;          // exact-zero diagonal
            dsum[r] += dd;
        }
    }
    float* out = (matId == 0) ? Demb : ((matId == 1) ? Dse : savg);
    const float mul = (matId == 2) ? (1.0f / (float)M) : 1.0f;
#pragma unroll
    for (int r = 0; r < 8; ++r) {
        int row = bi * 16 + r + 8 * half;
        out[row * N + rowB] = dsum[r] * mul;
    }
}

// --- 3a. row means (dm, sm) --------------------------------------------------
__global__ void rowsum_kernel(const float* __restrict__ Demb, const float* __restrict__ savg,
                              float* __restrict__ dm, float* __restrict__ sm) {
    int i = blockIdx.x * blockDim.x + threadIdx.x;
    if (i < N) {
        float s = 0.f;
        const float* r = Demb + (size_t)i * N;
        for (int j = 0; j < N; ++j) s += r[j];
        dm[i] = s / (float)(N - 1);
    } else if (i < 2 * N) {
        int k = i - N;
        float s = 0.f;
        const float* r = savg + (size_t)k * N;
        for (int j = 0; j < N; ++j) s += r[j];
        sm[k] = s / (float)(N - 1);
    }
}

// --- 3b. rkdd partial sums (64 blocks, fixed-order block reduce) -------------
__global__ void rkdd_kernel(const float* __restrict__ Demb, const float* __restrict__ savg,
                            const float* __restrict__ dm, const float* __restrict__ sm,
                            float* __restrict__ part) {
    __shared__ float red[256];
    const int total = N * N;
    float acc = 0.0f;
    for (int idx = blockIdx.x * blockDim.x + threadIdx.x; idx < total;
         idx += gridDim.x * blockDim.x) {
        int i = idx / N;
        float x = safe_div_f(Demb[idx], dm[i]);
        float y = safe_div_f(savg[idx], sm[i]);
        acc += huber_f(x, y);
    }
    red[threadIdx.x] = acc;
    __syncthreads();
    for (int s = blockDim.x >> 1; s > 0; s >>= 1) {
        if (threadIdx.x < (unsigned)s) red[threadIdx.x] += red[threadIdx.x + s];
        __syncthreads();
    }
    if (threadIdx.x == 0) part[blockIdx.x] = red[0];
}

// --- 4. miner: pos = argmin, neg = 16 smallest remaining ---------------------
__global__ void miner_kernel(const float* __restrict__ Dse, int* __restrict__ pos,
                             int* __restrict__ neg) {
    int i = blockIdx.x * blockDim.x + threadIdx.x;
    if (i >= N) return;
    const float* row = Dse + (size_t)i * N;
    float best = BIGV + 1.0f;
    int bidx = 0;
    for (int j = 0; j < N; ++j) {
        float v = (j == i) ? BIGV : row[j];
        if (v < best) { best = v; bidx = j; }   // first occurrence on ties (argmin)
    }
    pos[i] = bidx;
    float vals[NNEG];
    int idxs[NNEG];
#pragma unroll
    for (int t = 0; t < NNEG; ++t) { vals[t] = 3.0e38f; idxs[t] = 0; }
    for (int j = 0; j < N; ++j) {
        float v = (j == i || j == bidx) ? BIGV : row[j];
        if (v < vals[NNEG - 1]) {
            int p = NNEG - 1;
            while (p > 0 && vals[p - 1] > v) {
                vals[p] = vals[p - 1]; idxs[p] = idxs[p - 1]; --p;
            }
            vals[p] = v; idxs[p] = j;
        }
    }
    for (int t = 0; t < NNEG; ++t) neg[i * NNEG + t] = idxs[t];
}

// --- 5. rkda: one wave per (anchor, neg) pair; only the mined 6144 entries ---
__global__ void rkda_kernel(const float* __restrict__ emb, const float* __restrict__ src,
                            const int* __restrict__ pos, const int* __restrict__ neg,
                            float* __restrict__ part) {
    __shared__ float red[8];
    const int lane = threadIdx.x & 31;
    const int wv = threadIdx.x >> 5;
    const int pair = blockIdx.x * 8 + wv;        // 768 blocks * 8 waves == N*NNEG
    const int i = pair >> 4;
    const int p = pos[i];
    const int q = neg[pair];
    float uv[M + 1] = {0.f, 0.f, 0.f, 0.f};
    float uu[M + 1] = {0.f, 0.f, 0.f, 0.f};
    float vv[M + 1] = {0.f, 0.f, 0.f, 0.f};
    for (int t = lane; t < DD; t += 32) {
        float ei = emb[i * DD + t], ep = emb[p * DD + t], eq = emb[q * DD + t];
        float u = ep - ei, v = eq - ei;
        uv[0] += u * v; uu[0] += u * u; vv[0] += v * v;
#pragma unroll
        for (int s = 0; s < M; ++s) {
            float xi = src[(i * M + s) * DD + t];
            float xp = src[(p * M + s) * DD + t];
            float xq = src[(q * M + s) * DD + t];
            float us = xp - xi, vs = xq - xi;
            uv[1 + s] += us * vs; uu[1 + s] += us * us; vv[1 + s] += vs * vs;
        }
    }
#pragma unroll
    for (int j = 0; j <= M; ++j) {
        uv[j] = wred(uv[j]); uu[j] = wred(uu[j]); vv[j] = wred(vv[j]);
    }
    if (lane == 0) {
        float ang = uv[0] / (sqrtf(fmaxf(uu[0], 1e-24f)) * sqrtf(fmaxf(vv[0], 1e-24f)));
        float sang = 0.0f;
#pragma unroll
        for (int s = 1; s <= M; ++s)
            sang += uv[s] / (sqrtf(fmaxf(uu[s], 1e-24f)) * sqrtf(fmaxf(vv[s], 1e-24f)));
        sang *= (1.0f / (float)M);
        red[wv] = huber_f(ang, sang);
    }
    __syncthreads();
    if (threadIdx.x == 0) {
        float s = 0.f;
#pragma unroll
        for (int w = 0; w < 8; ++w) s += red[w];
        part[blockIdx.x] = s;
    }
}

// --- 6. deterministic fixed-order final combine ------------------------------
__global__ void final_kernel(const float* __restrict__ pdd, const float* __restrict__ pda,
                             float* __restrict__ out) {
    if (threadIdx.x == 0 && blockIdx.x == 0) {
        float s1 = 0.f;
        for (int b = 0; b < 64; ++b) s1 += pdd[b];
        float s2 = 0.f;
        for (int b = 0; b < 768; ++b) s2 += pda[b];
        float rkdd = s1 / (float)(N * N);
        float rkda = s2 / (float)(N * NNEG);
        out[0] = 1.0f * rkdd + 2.0f * rkda;
    }
}

extern "C" void kernel_launch(void* const* d_in, const int* in_sizes, int n_in,
                              void* d_out, int out_size, void* d_ws, size_t ws_size,
                              hipStream_t stream) {
    (void)in_sizes; (void)n_in; (void)out_size; (void)ws_size;
    const float* src = (const float*)d_in[0];   // [N, M, DD]
    const float* emb = (const float*)d_in[1];   // [N, DD]

    float* W = (float*)d_ws;
    float* se   = W; W += N * DD;       // mean source embedding
    float* sq   = W; W += 5 * N;        // squared norms: emb, se, src0..2
    float* Demb = W; W += N * N;
    float* Dse  = W; W += N * N;
    float* savg = W; W += N * N;
    float* dm   = W; W += N;
    float* sm   = W; W += N;
    float* pdd  = W; W += 64;           // rkdd block partials
    float* pda  = W; W += 768;          // rkda block partials
    int* pos = (int*)W; W += N;
    int* neg = (int*)W; W += N * NNEG;

    prep_kernel<<<3, 128, 0, stream>>>(emb, src, se, sq);

    dim3 gg(144, 3);                    // 144*4 waves = 576 tiles, y = matrix id
    gram_dist_kernel<<<gg, 128, 0, stream>>>(emb, src, se, sq, Demb, Dse, savg);

    rowsum_kernel<<<6, 128, 0, stream>>>(Demb, savg, dm, sm);
    rkdd_kernel<<<64, 256, 0, stream>>>(Demb, savg, dm, sm, pdd);
    miner_kernel<<<3, 128, 0, stream>>>(Dse, pos, neg);
    rkda_kernel<<<768, 256, 0, stream>>>(emb, src, pos, neg, pda);
    final_kernel<<<1, 32, 0, stream>>>(pdd, pda, (float*)d_out);
}